// HeteroGNNBA_14551349198940
// MI455X (gfx1250) — compile-verified
//
#include <hip/hip_runtime.h>
#include <hip/hip_bf16.h>

// ---------------------------------------------------------------------------
// HeteroGNN (SAGEConv ring) for MI455X / gfx1250, wave32.
// Dead-code-eliminated: output depends only on passes (l0,t1),(l0,t2),(l1,t2).
// GEMMs use V_WMMA_F32_16X16X4_F32 (f32 A/B, f32 accum) to match the f32
// reference exactly while still running on the CDNA5 matrix pipe.
// ---------------------------------------------------------------------------

#define DIMF 64
#define WAVES_PER_BLOCK 8

typedef __attribute__((ext_vector_type(2))) float v2f;
typedef __attribute__((ext_vector_type(8))) float v8f;

// ---------------- zero scratch (acc + cnt contiguous) ----------------------
__global__ void hg_zero_kernel(float* __restrict__ p, int n) {
    int i = blockIdx.x * blockDim.x + threadIdx.x;
    if (i < n) p[i] = 0.0f;
}

// ---------------- scatter-add of src features + degree count ---------------
// 16 threads per edge, each owns one float4 chunk of the 64-float row.
__global__ void hg_scatter_kernel(const long long* __restrict__ e,  // [2,E] int64
                                  const float* __restrict__ xsrc,   // [N,64]
                                  float* __restrict__ acc,          // [N,64]
                                  float* __restrict__ cnt,          // [N]
                                  int E) {
    int tid  = blockIdx.x * blockDim.x + threadIdx.x;
    int edge = tid >> 4;
    int f4   = tid & 15;
    if (edge >= E) return;
    long long s = e[edge];
    long long d = e[(long long)E + edge];
    float4 v = ((const float4*)(xsrc + s * DIMF))[f4];
    float* dst = acc + d * DIMF + f4 * 4;
    atomicAdd(dst + 0, v.x);
    atomicAdd(dst + 1, v.y);
    atomicAdd(dst + 2, v.z);
    atomicAdd(dst + 3, v.w);
    if (f4 == 0) atomicAdd(cnt + d, 1.0f);
}

// ---------------- fused SAGE layer: (acc/cnt)@Wl + bl + xroot@Wr, lrelu ----
// One wave = one 16-node tile. WMMA f32 16x16x4, K-loop of 16, 4 N-blocks.
// A layout (ISA 16x4 f32): lanes 0-15 -> rows M=0..15, VGPR0/1 = K {0,1};
// lanes 16-31 -> same rows, K {2,3}. B assumed symmetric (VGPR0/1 = rows
// K{0,1} for lanes 0-15 / K{2,3} for lanes 16-31, N = lane&15).
__global__ void __launch_bounds__(256)
hg_sage_wmma_kernel(const float* __restrict__ acc,
                    const float* __restrict__ cnt,
                    const float* __restrict__ xroot,
                    const float* __restrict__ Wl,    // [64,64]
                    const float* __restrict__ bias,  // [64]
                    const float* __restrict__ Wr,    // [64,64]
                    float* __restrict__ out,         // [N,64]
                    int nNodes) {
    __shared__ float sWl[DIMF * DIMF];
    __shared__ float sWr[DIMF * DIMF];

    int tid = threadIdx.x;
    for (int i = tid; i < DIMF * DIMF; i += 256) {
        sWl[i] = Wl[i];
        sWr[i] = Wr[i];
    }
    __syncthreads();

    int wave = tid >> 5;
    int lane = tid & 31;
    int hi   = lane >> 4;   // selects K pair {2,3} half
    int m    = lane & 15;   // A row within tile / B,C column within block
    int base = (blockIdx.x * WAVES_PER_BLOCK + wave) * 16;
    if (base >= nNodes) return;   // wave-uniform: EXEC stays all-ones for WMMA

    int node = base + m;
    float rcp = 1.0f / fmaxf(cnt[node], 1.0f);   // mean as row-scale of A

    // Preload this lane's A fragments for all 16 K-steps (reused 4x over N).
    v2f amean[16];
    v2f xr[16];
    const float* arow = acc   + (size_t)node * DIMF;
    const float* xrow = xroot + (size_t)node * DIMF;
#pragma unroll
    for (int s = 0; s < 16; ++s) {
        int kb = 4 * s + 2 * hi;
        v2f a; a.x = arow[kb] * rcp; a.y = arow[kb + 1] * rcp;
        amean[s] = a;
        v2f x; x.x = xrow[kb];       x.y = xrow[kb + 1];
        xr[s] = x;
    }

#pragma unroll
    for (int nb = 0; nb < 4; ++nb) {
        int n = nb * 16 + m;
        float bv = bias[n];                      // bias broadcast per column
        v8f c = { bv, bv, bv, bv, bv, bv, bv, bv };
#pragma unroll
        for (int s = 0; s < 16; ++s) {           // mean @ Wl
            int kb = 4 * s + 2 * hi;
            v2f b; b.x = sWl[kb * DIMF + n]; b.y = sWl[(kb + 1) * DIMF + n];
            c = __builtin_amdgcn_wmma_f32_16x16x4_f32(
                    false, amean[s], false, b, (short)0, c, false, false);
        }
#pragma unroll
        for (int s = 0; s < 16; ++s) {           // + xroot @ Wr
            int kb = 4 * s + 2 * hi;
            v2f b; b.x = sWr[kb * DIMF + n]; b.y = sWr[(kb + 1) * DIMF + n];
            c = __builtin_amdgcn_wmma_f32_16x16x4_f32(
                    false, xr[s], false, b, (short)0, c, false, false);
        }
        // C/D layout: VGPR v -> row v (lanes 0-15) or v+8 (lanes 16-31)
#pragma unroll
        for (int v = 0; v < 8; ++v) {
            float val = c[v];
            val = (val > 0.0f) ? val : 0.01f * val;   // leaky_relu fused
            out[(size_t)(base + v + 8 * hi) * DIMF + n] = val;
        }
    }
}

// ---------------- final Linear: [N,64] @ [64,2] + b ------------------------
__global__ void hg_final_linear_kernel(const float* __restrict__ z,
                                       const float* __restrict__ w,  // [64,2]
                                       const float* __restrict__ b,  // [2]
                                       float* __restrict__ out,      // [N,2]
                                       int n) {
    int i = blockIdx.x * blockDim.x + threadIdx.x;
    if (i >= n) return;
    const float* row = z + (size_t)i * DIMF;
    float a0 = b[0], a1 = b[1];
#pragma unroll 8
    for (int k = 0; k < DIMF; ++k) {
        float v = row[k];
        a0 += v * w[2 * k];
        a1 += v * w[2 * k + 1];
    }
    out[2 * (size_t)i]     = a0;
    out[2 * (size_t)i + 1] = a1;
}

// ---------------------------------------------------------------------------
extern "C" void kernel_launch(void* const* d_in, const int* in_sizes, int n_in,
                              void* d_out, int out_size, void* d_ws, size_t ws_size,
                              hipStream_t stream) {
    const float*     x1 = (const float*)d_in[1];
    const float*     x2 = (const float*)d_in[2];
    const float*     x3 = (const float*)d_in[3];
    const long long* e1 = (const long long*)d_in[5];   // edge type (1,2)
    const long long* e2 = (const long long*)d_in[6];   // edge type (2,3)
    const float*     Wl = (const float*)d_in[8];       // [2,4,64,64]
    const float*     bl = (const float*)d_in[9];       // [2,4,64]
    const float*     Wr = (const float*)d_in[10];      // [2,4,64,64]
    const float*     lw = (const float*)d_in[11];      // [64,2]
    const float*     lb = (const float*)d_in[12];      // [2]
    float*           out = (float*)d_out;

    const int N = in_sizes[0] / DIMF;   // 100000
    const int E = in_sizes[4] / 2;      // 800000

    // Workspace layout (floats): y2 | y3 | z3 | acc | cnt
    float* ws  = (float*)d_ws;
    float* y2  = ws;
    float* y3  = ws + (size_t)1 * N * DIMF;
    float* z3  = ws + (size_t)2 * N * DIMF;
    float* acc = ws + (size_t)3 * N * DIMF;
    float* cnt = ws + (size_t)4 * N * DIMF;   // contiguous after acc

    const int zeroN     = N * DIMF + N;
    const int zeroGrid  = (zeroN + 255) / 256;
    const int scatGrid  = (E * 16 + 255) / 256;
    const int sageGrid  = (N + WAVES_PER_BLOCK * 16 - 1) / (WAVES_PER_BLOCK * 16);
    const int linGrid   = (N + 255) / 256;

    const int WMAT = DIMF * DIMF;   // 4096 floats per weight matrix

    // ---- Pass A: layer 0, edge type 1 (1->2): y2 ----
    hg_zero_kernel<<<zeroGrid, 256, 0, stream>>>(acc, zeroN);
    hg_scatter_kernel<<<scatGrid, 256, 0, stream>>>(e1, x1, acc, cnt, E);
    hg_sage_wmma_kernel<<<sageGrid, 256, 0, stream>>>(
        acc, cnt, x2, Wl + 1 * WMAT, bl + 1 * DIMF, Wr + 1 * WMAT, y2, N);

    // ---- Pass B: layer 0, edge type 2 (2->3): y3 ----
    hg_zero_kernel<<<zeroGrid, 256, 0, stream>>>(acc, zeroN);
    hg_scatter_kernel<<<scatGrid, 256, 0, stream>>>(e2, x2, acc, cnt, E);
    hg_sage_wmma_kernel<<<sageGrid, 256, 0, stream>>>(
        acc, cnt, x3, Wl + 2 * WMAT, bl + 2 * DIMF, Wr + 2 * WMAT, y3, N);

    // ---- Pass C: layer 1, edge type 2 (2->3): z3 ----
    hg_zero_kernel<<<zeroGrid, 256, 0, stream>>>(acc, zeroN);
    hg_scatter_kernel<<<scatGrid, 256, 0, stream>>>(e2, y2, acc, cnt, E);
    hg_sage_wmma_kernel<<<sageGrid, 256, 0, stream>>>(
        acc, cnt, y3, Wl + 6 * WMAT, bl + 6 * DIMF, Wr + 6 * WMAT, z3, N);

    // ---- Final linear on node type 3 ----
    hg_final_linear_kernel<<<linGrid, 256, 0, stream>>>(z3, lw, lb, out, N);
}